// GCN_52458730553950
// MI455X (gfx1250) — compile-verified
//
#include <hip/hip_runtime.h>
#include <hip/hip_bf16.h>

// ---------------------------------------------------------------------------
// GCN (3x GCNConv + mean-pool + FC) for MI455X / gfx1250, wave32.
//  * GEMMs: V_WMMA_F32_16X16X4_F32 (exact fp32), B-operand served from LDS.
//  * W (64KB) staged into LDS once per block by the Tensor Data Mover
//    (tensor_load_to_lds) with hardware row padding to a 136-float stride so
//    the two wave halves hit disjoint LDS bank sets (2*136 mod 64 == 16).
//  * Scatter: one wave32 per edge (32 lanes x float4 = full 512B row) ->
//    coalesced gather + coalesced f32 atomics, working set lives in 192MB L2.
// ---------------------------------------------------------------------------

typedef float v2f  __attribute__((ext_vector_type(2)));
typedef float v8f  __attribute__((ext_vector_type(8)));
typedef unsigned int u32x4 __attribute__((ext_vector_type(4)));
typedef int   i32x4 __attribute__((ext_vector_type(4)));
typedef int   i32x8 __attribute__((ext_vector_type(8)));

#define NH 128          // NIN == NHID == 128
#define WSTRIDE 136     // LDS row stride in floats (bank-conflict-free)

// ---------------- utility: fill ----------------
__global__ void k_fill_f32(float* __restrict__ p, float v, long long n) {
  long long i = (long long)blockIdx.x * blockDim.x + threadIdx.x;
  if (i < n) p[i] = v;
}

// ---------------- degree / norm ----------------
__global__ void k_deg_accum(const long long* __restrict__ dst, float* __restrict__ deg,
                            long long E) {
  long long e = (long long)blockIdx.x * blockDim.x + threadIdx.x;
  if (e < E) atomicAdd(&deg[(int)dst[e]], 1.0f);
}

__global__ void k_deg_isqrt(float* __restrict__ deg, int N) {
  int i = blockIdx.x * blockDim.x + threadIdx.x;
  if (i < N) deg[i] = rsqrtf(deg[i]);   // deg >= 1 always (self loop pre-filled)
}

// ---------------- WMMA GEMM: H[N x 128] = X[N x 128] @ W[128 x 128] ----------
// 256 threads = 8 waves per block; wave w computes the 16x16 tile at
// (rows blockIdx.x*16 .. +15, cols w*16 .. +15). K loop: 32 x wmma_f32_16x16x4.
// W is staged to LDS once per block via the Tensor Data Mover.
__global__ __launch_bounds__(256) void k_gemm_wmma(
    const float* __restrict__ X, const float* __restrict__ W,
    float* __restrict__ H, int N) {
  __shared__ float lds_w[WSTRIDE * NH];   // 136*128*4 = 69632 B

#if defined(__gfx1250__) && __has_builtin(__builtin_amdgcn_tensor_load_to_lds)
  // ---- TDM: DMA the whole 128x128 f32 W into LDS with 8-DWORD row padding --
  if ((threadIdx.x >> 5) == 0) {                 // wave 0 only (uniform branch)
    const unsigned long long ga = (unsigned long long)W;
    const unsigned int lds_off  = (unsigned int)(unsigned long long)(&lds_w[0]);
    // D# group 0: count=1 | lds_addr | global_addr[56:0] | type=2
    u32x4 g0 = { 1u,
                 lds_off,
                 (unsigned int)ga,
                 (unsigned int)((ga >> 32) & 0x1FFFFFFu) | (2u << 30) };
    // D# group 1: data_size=4B(code 2), pad_enable, pad_interval=128dw(code 6),
    //             pad_amount=8dw(code 7); tensor 128x128, tile 128x128, stride 128
    i32x8 g1 = { (int)((2u << 16) | (1u << 20) | (6u << 22) | (7u << 25)),
                 (int)(128u << 16),   // tensor_dim0[15:0]=128 (bits 63:48)
                 (int)(128u << 16),   // tensor_dim0 hi=0 | tensor_dim1 lo=128
                 (int)(128u << 16),   // tensor_dim1 hi=0 | tile_dim0=128
                 128,                 // tile_dim1=128 | tile_dim2=0
                 128,                 // tensor_dim0_stride = 128 elements
                 0, 0 };
    i32x4 gz = { 0, 0, 0, 0 };        // groups 2/3 unused (2-D tensor)
#if __clang_major__ >= 23
    i32x8 gz8 = { 0, 0, 0, 0, 0, 0, 0, 0 };
    __builtin_amdgcn_tensor_load_to_lds(g0, g1, gz, gz, gz8, 0);
#else
    __builtin_amdgcn_tensor_load_to_lds(g0, g1, gz, gz, 0);
#endif
    __builtin_amdgcn_s_wait_tensorcnt(0);
  }
  __syncthreads();
#else
  // ---- fallback: cooperative staging ----
  for (int i = threadIdx.x; i < NH * NH; i += blockDim.x) {
    const int r = i >> 7, c = i & (NH - 1);
    lds_w[r * WSTRIDE + c] = W[i];
  }
  __syncthreads();
#endif

  const int wave = threadIdx.x >> 5;
  const int lane = threadIdx.x & 31;
  const int row0 = blockIdx.x << 4;            // 16-row stripe (N % 16 == 0)
  const int col0 = wave << 4;                  // 8 waves * 16 = 128 cols
  const int m    = lane & 15;                  // A: row index, B: col index
  const int kh   = lane >> 4;                  // lane-half selects K pair

  const float* __restrict__ xrow = X + (long long)(row0 + m) * NH;
  const float* wl = &lds_w[col0 + m];
  v8f acc = {};
#pragma unroll
  for (int k = 0; k < NH; k += 4) {
    const int ka = k + (kh << 1);              // this lane-half's K base
    v2f a, b;
    a.x = xrow[ka];                            // one global_load_b64 per wmma
    a.y = xrow[ka + 1];
    b.x = wl[ka * WSTRIDE];                    // ds_load_b32, conflict-free
    b.y = wl[(ka + 1) * WSTRIDE];
    // 8 args: (neg_a, A, neg_b, B, c_mod, C, reuse_a, reuse_b)
    acc = __builtin_amdgcn_wmma_f32_16x16x4_f32(
        false, a, false, b, (short)0, acc, false, false);
  }
#pragma unroll
  for (int r = 0; r < 8; ++r) {
    const int mr = r + (kh << 3);              // C/D layout: VGPR r -> M=r / r+8
    H[(long long)(row0 + mr) * NH + col0 + m] = acc[r];
  }
}

// ---------------- edge scatter: accum[dst] += xw[src] * norm ----------------
// One wave32 per edge: lane handles float4 chunk `lane` of the 128-float row.
__global__ void k_scatter_edges(const float* __restrict__ xw,
                                const long long* __restrict__ src,
                                const long long* __restrict__ dst,
                                const float* __restrict__ disq,
                                float* __restrict__ accum, long long E) {
  long long t = (long long)blockIdx.x * blockDim.x + threadIdx.x;
  if (t >= E * 32) return;
  const long long e = t >> 5;
  const int chunk   = (int)(t & 31);
  const int s = (int)src[e];
  const int d = (int)dst[e];
  const float nrm = disq[s] * disq[d];
  const float4 v = ((const float4*)(xw + (long long)s * NH))[chunk];
  float* out = accum + (long long)d * NH + chunk * 4;
  atomicAdd(out + 0, v.x * nrm);
  atomicAdd(out + 1, v.y * nrm);
  atomicAdd(out + 2, v.z * nrm);
  atomicAdd(out + 3, v.w * nrm);
}

// ---------------- epilogue: h = relu(accum + xw * disq^2 (self loop) + b) ----
__global__ void k_epilogue(const float* __restrict__ accum,
                           const float* __restrict__ xw,
                           const float* __restrict__ disq,
                           const float* __restrict__ bias,
                           float* __restrict__ h, long long total) {
  long long idx = (long long)blockIdx.x * blockDim.x + threadIdx.x;
  if (idx >= total) return;
  const int node = (int)(idx >> 7);            // / 128
  const int c    = (int)(idx & (NH - 1));
  const float ds = disq[node];
  const float v  = accum[idx] + xw[idx] * ds * ds + bias[c];
  h[idx] = fmaxf(v, 0.0f);
}

// ---------------- pooling: sums[g] += h[node]; cnts[g] += 1 ------------------
__global__ void k_pool(const float* __restrict__ h,
                       const long long* __restrict__ batch,
                       float* __restrict__ sums, float* __restrict__ cnts,
                       long long Nnodes) {
  long long t = (long long)blockIdx.x * blockDim.x + threadIdx.x;
  if (t >= Nnodes * 32) return;
  const long long node = t >> 5;
  const int chunk      = (int)(t & 31);
  const int g = (int)batch[node];
  const float4 v = ((const float4*)(h + node * NH))[chunk];
  float* out = sums + (long long)g * NH + chunk * 4;
  atomicAdd(out + 0, v.x);
  atomicAdd(out + 1, v.y);
  atomicAdd(out + 2, v.z);
  atomicAdd(out + 3, v.w);
  if (chunk == 0) atomicAdd(&cnts[g], 1.0f);
}

// ---------------- final FC: out[g] = (sums[g]/max(cnt,1)) . Wfc + bfc --------
__global__ void k_fc(const float* __restrict__ sums, const float* __restrict__ cnts,
                     const float* __restrict__ Wfc, const float* __restrict__ bfc,
                     float* __restrict__ out, int G) {
  int g = blockIdx.x * blockDim.x + threadIdx.x;
  if (g >= G) return;
  const float inv = 1.0f / fmaxf(cnts[g], 1.0f);
  const float* row = sums + (long long)g * NH;
  float acc = 0.0f;
#pragma unroll 8
  for (int k = 0; k < NH; ++k) acc += row[k] * Wfc[k];
  out[g] = acc * inv + bfc[0];
}

// ---------------------------------------------------------------------------
extern "C" void kernel_launch(void* const* d_in, const int* in_sizes, int n_in,
                              void* d_out, int out_size, void* d_ws, size_t ws_size,
                              hipStream_t stream) {
  const float*     x     = (const float*)d_in[0];
  const long long* ei    = (const long long*)d_in[1];   // int64 [2, E]
  const long long* batch = (const long long*)d_in[2];   // int64 [N]
  const float* W1  = (const float*)d_in[3];
  const float* b1  = (const float*)d_in[4];
  const float* W2  = (const float*)d_in[5];
  const float* b2  = (const float*)d_in[6];
  const float* W3  = (const float*)d_in[7];
  const float* b3  = (const float*)d_in[8];
  const float* Wfc = (const float*)d_in[9];
  const float* bfc = (const float*)d_in[10];
  float* out = (float*)d_out;

  const long long E = in_sizes[1] / 2;
  const int  N      = in_sizes[2];           // 100000 (multiple of 16)
  const int  G      = out_size;              // 1024
  const long long* src = ei;
  const long long* dst = ei + E;

  // ---- carve workspace (512B-aligned slabs) ----
  char*  ws  = (char*)d_ws;
  size_t off = 0;
  auto carve = [&](size_t bytes) -> float* {
    float* p = (float*)(ws + off);
    off = (off + bytes + 511) & ~(size_t)511;
    return p;
  };
  const long long feat = (long long)N * NH;
  float* disq = carve((size_t)N * sizeof(float));
  float* bufA = carve((size_t)feat * sizeof(float));
  float* bufB = carve((size_t)feat * sizeof(float));
  float* bufC = carve((size_t)feat * sizeof(float));
  float* sums = carve((size_t)G * NH * sizeof(float));
  float* cnts = carve((size_t)G * sizeof(float));
  (void)ws_size;

  const int TB = 256;
  auto blocks = [&](long long n) { return (unsigned)((n + TB - 1) / TB); };
  const unsigned gemm_grid  = (unsigned)(N / 16);
  const long long edge_work = E * 32;
  const long long pool_work = (long long)N * 32;

  // ---- degrees -> deg^{-1/2} (self-loop folded in as initial 1.0) ----
  k_fill_f32<<<blocks(N), TB, 0, stream>>>(disq, 1.0f, N);
  k_deg_accum<<<blocks(E), TB, 0, stream>>>(dst, disq, E);
  k_deg_isqrt<<<blocks(N), TB, 0, stream>>>(disq, N);

  // ---- layer 1: x -> bufA(xw) -> bufB(h1) ----
  k_gemm_wmma<<<gemm_grid, TB, 0, stream>>>(x, W1, bufA, N);
  k_fill_f32<<<blocks(feat), TB, 0, stream>>>(bufB, 0.0f, feat);
  k_scatter_edges<<<blocks(edge_work), TB, 0, stream>>>(bufA, src, dst, disq, bufB, E);
  k_epilogue<<<blocks(feat), TB, 0, stream>>>(bufB, bufA, disq, b1, bufB, feat);

  // ---- layer 2: bufB(h1) -> bufA(xw) -> bufC(h2) ----
  k_gemm_wmma<<<gemm_grid, TB, 0, stream>>>(bufB, W2, bufA, N);
  k_fill_f32<<<blocks(feat), TB, 0, stream>>>(bufC, 0.0f, feat);
  k_scatter_edges<<<blocks(edge_work), TB, 0, stream>>>(bufA, src, dst, disq, bufC, E);
  k_epilogue<<<blocks(feat), TB, 0, stream>>>(bufC, bufA, disq, b2, bufC, feat);

  // ---- layer 3: bufC(h2) -> bufA(xw) -> bufB(h3) ----
  k_gemm_wmma<<<gemm_grid, TB, 0, stream>>>(bufC, W3, bufA, N);
  k_fill_f32<<<blocks(feat), TB, 0, stream>>>(bufB, 0.0f, feat);
  k_scatter_edges<<<blocks(edge_work), TB, 0, stream>>>(bufA, src, dst, disq, bufB, E);
  k_epilogue<<<blocks(feat), TB, 0, stream>>>(bufB, bufA, disq, b3, bufB, feat);

  // ---- mean pool + FC ----
  k_fill_f32<<<blocks((long long)G * NH), TB, 0, stream>>>(sums, 0.0f, (long long)G * NH);
  k_fill_f32<<<blocks(G), TB, 0, stream>>>(cnts, 0.0f, G);
  k_pool<<<blocks(pool_work), TB, 0, stream>>>(bufB, batch, sums, cnts, N);
  k_fc<<<blocks(G), TB, 0, stream>>>(sums, cnts, Wfc, bfc, out, G);
}